// Invert_68779606278977
// MI455X (gfx1250) — compile-verified
//
#include <hip/hip_runtime.h>

typedef float f4 __attribute__((ext_vector_type(4)));

#define TPB 256
#define NBLK 2048

__device__ __forceinline__ unsigned row_of(unsigned e, unsigned n, int shift) {
    return (shift >= 0) ? (e >> shift) : (e / n);
}

// Main streaming kernel: partial[b] = sum over its elements of (X[row]*A[row,col])^2
// Diagonal handled in the finalizer: (xa-1)^2 = (xa)^2 - 2xa + 1.
// Unrolled x4: four independent non-temporal b128 loads in flight per wave
// to reach the MLP needed for 23.3 TB/s HBM streaming.
__global__ __launch_bounds__(TPB) void rowscale_sq_partial(
    const float* __restrict__ X, const float* __restrict__ A,
    float* __restrict__ partials, unsigned nv4, unsigned n, int shift)
{
    const unsigned tid    = blockIdx.x * blockDim.x + threadIdx.x;
    const unsigned stride = gridDim.x * blockDim.x;
    const f4* __restrict__ A4 = (const f4*)A;

    f4 acc0 = {0.f,0.f,0.f,0.f}, acc1 = acc0, acc2 = acc0, acc3 = acc0;

    unsigned v = tid;
    const unsigned last = nv4 - 1;
    for (; v + 3u * stride < nv4; v += 4u * stride) {
        const unsigned i0 = v, i1 = v + stride, i2 = v + 2u*stride, i3 = v + 3u*stride;

        // Speculative prefetch one unrolled iteration ahead (clamped, branch-free).
        __builtin_prefetch((const void*)(A4 + min(i0 + 4u*stride, last)), 0, 1);
        __builtin_prefetch((const void*)(A4 + min(i1 + 4u*stride, last)), 0, 1);
        __builtin_prefetch((const void*)(A4 + min(i2 + 4u*stride, last)), 0, 1);
        __builtin_prefetch((const void*)(A4 + min(i3 + 4u*stride, last)), 0, 1);

        // Four independent NT 128-bit loads issued before any use.
        f4 a0 = __builtin_nontemporal_load(A4 + i0);
        f4 a1 = __builtin_nontemporal_load(A4 + i1);
        f4 a2 = __builtin_nontemporal_load(A4 + i2);
        f4 a3 = __builtin_nontemporal_load(A4 + i3);

        const float x0 = X[row_of(i0 << 2, n, shift)];
        const float x1 = X[row_of(i1 << 2, n, shift)];
        const float x2 = X[row_of(i2 << 2, n, shift)];
        const float x3 = X[row_of(i3 << 2, n, shift)];

        f4 t0 = a0 * x0; acc0 += t0 * t0;
        f4 t1 = a1 * x1; acc1 += t1 * t1;
        f4 t2 = a2 * x2; acc2 += t2 * t2;
        f4 t3 = a3 * x3; acc3 += t3 * t3;
    }
    // Tail (only taken when the ws-size fallback shrinks the grid).
    for (; v < nv4; v += stride) {
        f4 a = __builtin_nontemporal_load(A4 + v);
        float x = X[row_of(v << 2, n, shift)];
        f4 t = a * x;
        acc0 += t * t;
    }

    f4 accv = (acc0 + acc1) + (acc2 + acc3);
    float acc = (accv.x + accv.y) + (accv.z + accv.w);

    // wave32 reduction (CDNA5 is wave32-only)
    for (int off = 16; off > 0; off >>= 1)
        acc += __shfl_down(acc, off, 32);

    __shared__ float wsum[TPB / 32];
    const int lane = threadIdx.x & 31;
    const int wave = threadIdx.x >> 5;
    if (lane == 0) wsum[wave] = acc;
    __syncthreads();
    if (wave == 0) {
        float s = (lane < (int)(blockDim.x >> 5)) ? wsum[lane] : 0.f;
        for (int off = 4; off > 0; off >>= 1)
            s += __shfl_down(s, off, 32);
        if (lane == 0) partials[blockIdx.x] = s;
    }
}

// Single-block finalizer: fp64 sum of block partials + diagonal correction,
// result = sum((xa)^2) - 2*sum(x_i * a_ii) + n
__global__ __launch_bounds__(TPB) void finalize_kernel(
    const float* __restrict__ partials, int nblocks,
    const float* __restrict__ X, const float* __restrict__ A,
    unsigned n, float* __restrict__ out)
{
    double s = 0.0;
    for (int i = threadIdx.x; i < nblocks; i += blockDim.x)
        s += (double)partials[i];

    double dsum = 0.0;
    for (unsigned i = threadIdx.x; i < n; i += blockDim.x)
        dsum += (double)X[i] * (double)A[(size_t)i * n + i];

    __shared__ double sh[TPB];
    sh[threadIdx.x] = s - 2.0 * dsum;
    __syncthreads();
    for (int off = TPB / 2; off > 0; off >>= 1) {
        if ((int)threadIdx.x < off) sh[threadIdx.x] += sh[threadIdx.x + off];
        __syncthreads();
    }
    if (threadIdx.x == 0)
        out[0] = (float)(sh[0] + (double)n);
}

extern "C" void kernel_launch(void* const* d_in, const int* in_sizes, int n_in,
                              void* d_out, int out_size, void* d_ws, size_t ws_size,
                              hipStream_t stream) {
    const float* X = (const float*)d_in[0];
    const float* A = (const float*)d_in[1];
    float* out = (float*)d_out;

    const unsigned n   = (unsigned)in_sizes[0];           // 8192
    const unsigned nv4 = (unsigned)(((size_t)n * n) >> 2);

    // Power-of-two fast path for row = element_index / n.
    int shift = -1;
    if ((n & (n - 1u)) == 0u) {
        shift = 0;
        while ((1u << shift) < n) ++shift;
    }

    int nblocks = NBLK;
    if (ws_size < (size_t)nblocks * sizeof(float)) {
        nblocks = (int)(ws_size / sizeof(float));
        if (nblocks < 1) nblocks = 1;
    }
    float* partials = (float*)d_ws;

    rowscale_sq_partial<<<nblocks, TPB, 0, stream>>>(X, A, partials, nv4, n, shift);
    finalize_kernel<<<1, TPB, 0, stream>>>(partials, nblocks, X, A, n, out);
}